// LabelAssignment_53077205844453
// MI455X (gfx1250) — compile-verified
//
#include <hip/hip_runtime.h>

typedef __attribute__((ext_vector_type(2))) float v2f;
typedef __attribute__((ext_vector_type(8))) float v8f;

#define TPB 256
#define KMAX 64
#define IOU_THR 0.6f

// Exact Sutherland-Hodgman intersection area of two CCW rects (reference semantics).
__device__ float rect_inter_area(const float ax[4], const float ay[4],
                                 const float bx[4], const float by[4]) {
  float px[8], py[8], qx[8], qy[8];
  int n = 4;
#pragma unroll
  for (int i = 0; i < 4; ++i) { px[i] = ax[i]; py[i] = ay[i]; }
  for (int e = 0; e < 4; ++e) {
    float e0x = bx[e], e0y = by[e];
    float e1x = bx[(e + 1) & 3], e1y = by[(e + 1) & 3];
    float evx = e1x - e0x, evy = e1y - e0y;
    int m = 0;
    for (int i = 0; i < n; ++i) {
      int j = (i + 1 == n) ? 0 : i + 1;
      float di = evx * (py[i] - e0y) - evy * (px[i] - e0x);
      float dj = evx * (py[j] - e0y) - evy * (px[j] - e0x);
      bool ii = di >= 0.f, ij = dj >= 0.f;
      if (ii != ij) {
        float den = di - dj;
        den = (fabsf(den) < 1e-12f) ? 1e-12f : den;
        float t = di / den;
        qx[m] = px[i] + t * (px[j] - px[i]);
        qy[m] = py[i] + t * (py[j] - py[i]);
        ++m;
      }
      if (ij) { qx[m] = px[j]; qy[m] = py[j]; ++m; }
    }
    n = m;
    for (int i = 0; i < n; ++i) { px[i] = qx[i]; py[i] = qy[i]; }
  }
  float area = 0.f;
  for (int i = 0; i < n; ++i) {
    int j = (i + 1 == n) ? 0 : i + 1;
    area += px[i] * py[j] - px[j] * py[i];
  }
  return 0.5f * fabsf(area);
}

__global__ __launch_bounds__(TPB) void iou_assign_kernel(
    const float* __restrict__ boxes, const float* __restrict__ gt,
    int* __restrict__ out, int N, int K) {
  __shared__ float graw[KMAX * 7];
  __shared__ float gcx[KMAX][4], gcy[KMAX][4];
  __shared__ float gz1[KMAX], gz2[KMAX], gvol[KMAX];
  __shared__ float ggx[KMAX], ggy[KMAX], gnb[KMAX], grb[KMAX];
  __shared__ float bxl[TPB], byl[TPB], nal[TPB], ral[TPB];
  __shared__ unsigned short pmask[TPB][4];

  const int tid = threadIdx.x;
  const int b = blockIdx.y;
  const int i = blockIdx.x * TPB + tid;
  const bool valid = (i < N);
  const int ic = valid ? i : (N - 1);

  const float* gbase = gt + (size_t)b * K * 7;

  // ---- CDNA5 async global -> LDS staging of raw gt boxes (ASYNCcnt path) ----
  for (int d = tid; d < K * 7; d += TPB) {
    unsigned ldsa = (unsigned)(unsigned long long)(&graw[d]);
    const float* ga = gbase + d;
    asm volatile("global_load_async_to_lds_b32 %0, %1, off"
                 :: "v"(ldsa), "v"(ga) : "memory");
  }
  asm volatile("s_wait_asynccnt 0x0" ::: "memory");
  __syncthreads();

  // ---- derive per-gt records once per block ----
  if (tid < K) {
    float x = graw[tid * 7 + 0], y = graw[tid * 7 + 1], z = graw[tid * 7 + 2];
    float dx = graw[tid * 7 + 3], dy = graw[tid * 7 + 4], dz = graw[tid * 7 + 5];
    float ang = graw[tid * 7 + 6];
    float c = cosf(ang), s = sinf(ang);
    float hx = 0.5f * dx, hy = 0.5f * dy;
    const float lx[4] = { hx, -hx, -hx, hx };
    const float ly[4] = { hy, hy, -hy, -hy };
#pragma unroll
    for (int q = 0; q < 4; ++q) {
      gcx[tid][q] = x + lx[q] * c - ly[q] * s;
      gcy[tid][q] = y + lx[q] * s + ly[q] * c;
    }
    gz1[tid] = z - 0.5f * dz; gz2[tid] = z + 0.5f * dz;
    gvol[tid] = dx * dy * dz;
    ggx[tid] = x; ggy[tid] = y;
    gnb[tid] = x * x + y * y;
    grb[tid] = 0.5f * sqrtf(dx * dx + dy * dy);
  }

  // ---- own box: corners, z range, volume; publish center/norm/radius ----
  const float* bp = boxes + ((size_t)b * N + ic) * 7;
  float x = bp[0], y = bp[1], z = bp[2];
  float dx = bp[3], dy = bp[4], dz = bp[5], ang = bp[6];
  float c = cosf(ang), s = sinf(ang);
  float ax[4], ay[4];
  {
    float hx = 0.5f * dx, hy = 0.5f * dy;
    const float lx[4] = { hx, -hx, -hx, hx };
    const float ly[4] = { hy, hy, -hy, -hy };
#pragma unroll
    for (int q = 0; q < 4; ++q) {
      ax[q] = x + lx[q] * c - ly[q] * s;
      ay[q] = y + lx[q] * s + ly[q] * c;
    }
  }
  float az1 = z - 0.5f * dz, az2 = z + 0.5f * dz, avol = dx * dy * dz;
  bxl[tid] = x; byl[tid] = y;
  nal[tid] = x * x + y * y;
  ral[tid] = 0.5f * sqrtf(dx * dx + dy * dy);
  __syncthreads();

  // ---- WMMA center-distance prefilter: exact (circumradius) box-overlap bound ----
  const int lane = tid & 31;
  const int wave = tid >> 5;
  const bool lolane = (lane < 16);
  const int l15 = lane & 15;
  const int ntiles = (K + 15) / 16;
#pragma unroll
  for (int mt = 0; mt < 2; ++mt) {
    const int rowbase = wave * 32 + mt * 16;
    v2f A;                       // A 16x4 f32: lanes 0-15 K0,K1 = (x,y); lanes 16-31 K2,K3 = 0
    A.x = lolane ? bxl[rowbase + l15] : 0.f;
    A.y = lolane ? byl[rowbase + l15] : 0.f;
    for (int nt = 0; nt < ntiles; ++nt) {
      const int colbase = nt * 16;
      v2f Bm;                    // B 4x16 f32: lanes 0-15 rows K0,K1 = (gx,gy); lanes 16-31 = 0
      Bm.x = lolane ? ggx[colbase + l15] : 0.f;
      Bm.y = lolane ? ggy[colbase + l15] : 0.f;
      v8f Cz = {0.f, 0.f, 0.f, 0.f, 0.f, 0.f, 0.f, 0.f};
      v8f D = __builtin_amdgcn_wmma_f32_16x16x4_f32(
          false, A, false, Bm, (short)0, Cz, false, false);
      const int ncol = colbase + l15;
      const float nbn = gnb[ncol], rbn = grb[ncol];
#pragma unroll
      for (int r = 0; r < 8; ++r) {
        int row = rowbase + r + (lolane ? 0 : 8);  // D layout: vgpr r -> rows r / r+8
        float dist2 = nal[row] + nbn - 2.f * D[r];
        float thr = ral[row] + rbn;
        bool pass = dist2 <= thr * thr + 1e-3f;    // slack covers fp cancellation
        unsigned bal = __builtin_amdgcn_ballot_w32(pass);
        if (lane == 0) {
          pmask[rowbase + r][nt] = (unsigned short)(bal & 0xFFFFu);
          pmask[rowbase + r + 8][nt] = (unsigned short)(bal >> 16);
        }
      }
    }
  }
  __syncthreads();

  unsigned long long mask =
      (unsigned long long)pmask[tid][0] |
      ((unsigned long long)pmask[tid][1] << 16) |
      ((unsigned long long)pmask[tid][2] << 32) |
      ((unsigned long long)pmask[tid][3] << 48);

  // ---- full rotated IoU only for surviving pairs; first-max argmax semantics ----
  float best = -1.f;
  int bi = 0;
  for (int k = 0; k < K; ++k) {
    if ((mask >> k) & 1ull) {
      float bcx[4], bcy[4];
#pragma unroll
      for (int q = 0; q < 4; ++q) { bcx[q] = gcx[k][q]; bcy[q] = gcy[k][q]; }
      float area = rect_inter_area(ax, ay, bcx, bcy);
      float h = fmaxf(fminf(az2, gz2[k]) - fmaxf(az1, gz1[k]), 0.f);
      float inter = area * h;
      float iou = inter / fmaxf(avol + gvol[k] - inter, 1e-6f);
      if (iou > best) { best = iou; bi = k; }
    }
  }
  if (valid) out[(size_t)b * N + i] = (best < IOU_THR) ? -1 : bi;
}

extern "C" void kernel_launch(void* const* d_in, const int* in_sizes, int n_in,
                              void* d_out, int out_size, void* d_ws, size_t ws_size,
                              hipStream_t stream) {
  (void)n_in; (void)out_size; (void)d_ws; (void)ws_size;
  const float* tb = (const float*)d_in[0];
  const float* sb = (const float*)d_in[1];
  const float* gb = (const float*)d_in[2];
  const int B = 4;
  const int N = in_sizes[0] / (B * 7);
  const int M = in_sizes[1] / (B * 7);
  const int K = in_sizes[2] / (B * 7);
  if (N <= 0 || M <= 0 || K <= 0 || K > KMAX) return;

  int* out = (int*)d_out;
  dim3 blk(TPB);
  dim3 g1((N + TPB - 1) / TPB, B);
  dim3 g2((M + TPB - 1) / TPB, B);
  hipLaunchKernelGGL(iou_assign_kernel, g1, blk, 0, stream, tb, gb, out, N, K);
  hipLaunchKernelGGL(iou_assign_kernel, g2, blk, 0, stream, sb, gb,
                     out + (size_t)B * N, M, K);
}